// PNC_84327387890272
// MI455X (gfx1250) — compile-verified
//
#include <hip/hip_runtime.h>

// Problem constants (from reference)
#define V_   400000
#define D_   200
#define C_   45
#define B_   32
#define S_   128
#define KN_  12
#define KC_  10
#define NTOK (B_ * S_)   // 4096 tokens
#define KF   (5 * D_)    // 1000 = GEMM K

typedef __attribute__((ext_vector_type(2))) float v2f;
typedef __attribute__((ext_vector_type(4))) float v4f;
typedef __attribute__((ext_vector_type(8))) float v8f;

// ---------------------------------------------------------------------------
// Kernel 1: masked gather + average -> feat [NTOK, D_] (fp32)
// One block (2 waves) per token; lanes 0..49 each own a float4 slice of D=200.
// Mask/ids are uniform per block (scalar-cached); embed row reads are coalesced.
// ---------------------------------------------------------------------------
__global__ void __launch_bounds__(64) feat_gather_kernel(
    const int* __restrict__ ngram_ids, const int* __restrict__ ngram_mask,
    const int* __restrict__ ctx_ids,   const int* __restrict__ ctx_mask,
    const float* __restrict__ embed,   float* __restrict__ feat) {
  const int tok = blockIdx.x;          // 0..4095
  const int t   = threadIdx.x;         // 0..63
  const int d4  = t * 4;
  const bool act = (d4 < D_);
  const int dld = act ? d4 : (D_ - 4); // clamp so loads are always in-bounds

  const int* nid = ngram_ids  + tok * KN_;
  const int* nmk = ngram_mask + tok * KN_;
  const int* cid = ctx_ids    + tok * KC_;
  const int* cmk = ctx_mask   + tok * KC_;

  v4f acc = {0.f, 0.f, 0.f, 0.f};
  int cnt = 0;

#pragma unroll
  for (int j = 0; j < KN_; ++j) {
    const int m = nmk[j];
    cnt += m;
    const size_t base = (size_t)nid[j] * D_;
    v4f e = *reinterpret_cast<const v4f*>(embed + base + dld);
    acc += e * (float)m;               // mask by multiply (ids always valid)
  }
#pragma unroll
  for (int j = 0; j < KC_; ++j) {
    const int m = cmk[j];
    cnt += m;
    const size_t base = (size_t)cid[j] * D_;
    v4f e = *reinterpret_cast<const v4f*>(embed + base + dld);
    acc += e * (float)m;
  }

  const float scale = 1.0f / fmaxf((float)cnt, 1.0f);
  if (act) {
    *reinterpret_cast<v4f*>(feat + (size_t)tok * D_ + d4) = acc * scale;
  }
}

// ---------------------------------------------------------------------------
// Kernel 2: windowed GEMM via V_WMMA_F32_16X16X4_F32.
//   logit[m, n] = sum_{i=0..4} sum_{d=0..199} feat[b, s+i-2, d] * W[n, i*200+d] + bias[n]
// One wave32 per (16-row M tile, 16-col N tile). The window-5 shift is folded
// into the A-fragment address; out-of-window rows and n>=45 cols are
// value-masked (v_cndmask) so EXEC stays all-ones for WMMA.
//
// f32 WMMA fragment layout (ISA 7.12.2):
//   A: lane L holds (M = L&15, K = 2*(L>>4) + v), v in {0,1}  -> float2 load
//   B: lane L holds (K = 2*(L>>4) + v, N = L&15)              -> float2 load
//   D: vgpr r -> M = r + 8*(L>>4), N = L&15
// ---------------------------------------------------------------------------
__global__ void __launch_bounds__(32) window_gemm_wmma_kernel(
    const float* __restrict__ feat,  // [NTOK, D_]
    const float* __restrict__ W,     // [C_, KF] row-major
    const float* __restrict__ bias,  // [C_]
    float* __restrict__ out) {       // [NTOK, C_]
  const int mtile = blockIdx.x;      // 0..255 (16 tokens each; tile never crosses a batch)
  const int n0    = blockIdx.y * 16; // 0,16,32
  const int lane  = threadIdx.x;     // 0..31
  const int half  = lane >> 4;       // 0 | 1
  const int l15   = lane & 15;
  const int kkb   = half * 2;        // K sub-offset {0,2}

  const int m0 = mtile * 16;
  const int b0 = m0 / S_;
  const int s0 = m0 % S_;
  const int sA = s0 + l15;           // this lane's A-row sequence position

  // B-side (weights): column of W^T, masked past C_=45
  const int  col = n0 + l15;
  const bool vb  = (col < C_);
  const float bm = vb ? 1.0f : 0.0f;
  const float* wrow = W + (size_t)(vb ? col : 0) * KF + kkb;

  v8f acc = {0.f, 0.f, 0.f, 0.f, 0.f, 0.f, 0.f, 0.f};

  for (int i = 0; i < 5; ++i) {
    const int  sp = sA + i - 2;                 // shifted seq position
    const bool va = (sp >= 0) && (sp < S_);     // zero-pad window edges
    const float am = va ? 1.0f : 0.0f;
    const float* arow = feat + (size_t)(b0 * S_ + (va ? sp : 0)) * D_ + kkb;
    const float* wseg = wrow + i * D_;
#pragma unroll 5
    for (int d = 0; d < D_; d += 4) {           // 50 K-steps of 4 per window slot
      v2f a  = *reinterpret_cast<const v2f*>(arow + d);  // 8B-aligned
      v2f bv = *reinterpret_cast<const v2f*>(wseg + d);  // 8B-aligned
      a  *= am;
      bv *= bm;
      acc = __builtin_amdgcn_wmma_f32_16x16x4_f32(
          /*neg_a=*/false, a, /*neg_b=*/false, bv,
          /*c_mod=*/(short)0, acc, /*reuse_a=*/false, /*reuse_b=*/false);
    }
  }

  // Store D + bias; WMMA is done, divergence is now fine.
  if (vb) {
    const float bb   = bias[col];
    const int   mrow = m0 + half * 8;
#pragma unroll
    for (int r = 0; r < 8; ++r) {
      out[(size_t)(mrow + r) * C_ + col] = acc[r] + bb;
    }
  }
}

// ---------------------------------------------------------------------------
// Launch
// ---------------------------------------------------------------------------
extern "C" void kernel_launch(void* const* d_in, const int* in_sizes, int n_in,
                              void* d_out, int out_size, void* d_ws, size_t ws_size,
                              hipStream_t stream) {
  (void)in_sizes; (void)n_in; (void)out_size; (void)ws_size;

  const int*   ngram_ids  = (const int*)d_in[0];
  const int*   ngram_mask = (const int*)d_in[1];
  const int*   ctx_ids    = (const int*)d_in[2];
  const int*   ctx_mask   = (const int*)d_in[3];
  const float* embed      = (const float*)d_in[4];
  const float* W          = (const float*)d_in[5];
  const float* bias       = (const float*)d_in[6];
  float*       out        = (float*)d_out;

  float* feat = (float*)d_ws;  // NTOK * D_ * 4 = 3.2 MB scratch

  feat_gather_kernel<<<dim3(NTOK), dim3(64), 0, stream>>>(
      ngram_ids, ngram_mask, ctx_ids, ctx_mask, embed, feat);

  window_gemm_wmma_kernel<<<dim3(NTOK / 16, 3), dim3(32), 0, stream>>>(
      feat, W, bias, out);
}